// LightweightMHA_12249246728539
// MI455X (gfx1250) — compile-verified
//
#include <hip/hip_runtime.h>
#include <math.h>

typedef float v2f __attribute__((ext_vector_type(2)));
typedef float v8f __attribute__((ext_vector_type(8)));
typedef unsigned int u32x4 __attribute__((ext_vector_type(4)));
typedef int i32x4 __attribute__((ext_vector_type(4)));
typedef int i32x8 __attribute__((ext_vector_type(8)));

#define NB 1024
#define NC 64
#define NT 512
#define ND 32
#define NH 4
#define NDK 8
#define HS 36    // LDS row stride for 64x32 tiles (36*r mod 64 has period 16)
#define SS 66    // LDS row stride for 64x64 score tiles
#define WQS 40   // staged 32-wide weights: stride 40 (row pairs differ by 16 banks)
#define XCS 68   // staged x chunk: stride 68 (4*r mod 64 -> all banks once)
#define WOS 264  // staged W_out half: stride 264 (row pairs differ by 16 banks)

static constexpr float EPS = 1e-5f;

// ---- LDS arena (floats) ----
#define OFF_HO   0        // 64*36 = 2304   h / attention output o
#define OFF_Q    2304
#define OFF_K    4608
#define OFF_V    6912
#define OFF_SC   9216     // 64*66 = 4224   qkv weights (TDM), then scores
#define OFF_C    13440    // 16896 floats staging: P1 x/Win dbl-buf, then W_out halves
#define OFF_P1   30336    // 128
#define OFF_P2   30464    // 128
#define OFF_MU   30592    // 64
#define OFF_RS   30656    // 64
#define LDS_F    30720    // total floats = 122880 bytes

// staging sub-offsets (floats, relative to OFF_C)
#define XB0  0            // 64*68 = 4352
#define XB1  4352
#define WB0  8704         // 64*40 = 2560
#define WB1  11264
#define WO0  0            // 32*264 = 8448 (after P1)
#define WO1  8448

__device__ __forceinline__ v8f wmma_f32(v2f a, v2f b, v8f c) {
  return __builtin_amdgcn_wmma_f32_16x16x4_f32(false, a, false, b, (short)0, c,
                                               false, false);
}

// TDM: DMA a 2D fp32 tile (tile0 x tile1) from global into LDS with LDS padding.
__device__ __forceinline__ void tdm_load_2d_f32(uint32_t lds_byte_off,
                                                const void* gptr,
                                                uint32_t tdim0, uint32_t tdim1,
                                                uint32_t stride0,
                                                uint32_t tile0, uint32_t tile1,
                                                uint32_t g1flags) {
  const uint64_t ga = (uint64_t)(uintptr_t)gptr;
  u32x4 g0;
  g0.x = 1u;                                   // count=1 (valid user descriptor)
  g0.y = lds_byte_off;                         // lds_addr
  g0.z = (uint32_t)ga;                         // global_addr[31:0]
  g0.w = (uint32_t)((ga >> 32) & 0x01FFFFFFu)  // global_addr[56:32]
         | (2u << 30);                         // type=2 ("image")
  i32x8 g1;
  g1[0] = (int)g1flags;                        // mask=0 | data_size | pad cfg
  g1[1] = (int)((tdim0 & 0xFFFFu) << 16);
  g1[2] = (int)((tdim0 >> 16) | ((tdim1 & 0xFFFFu) << 16));
  g1[3] = (int)((tdim1 >> 16) | (tile0 << 16));
  g1[4] = (int)tile1;
  g1[5] = (int)stride0;
  g1[6] = 0;
  g1[7] = 0;
  const i32x4 z4 = {0, 0, 0, 0};
#if defined(__clang_major__) && (__clang_major__ >= 23)
  const i32x8 z8 = {0, 0, 0, 0, 0, 0, 0, 0};
  __builtin_amdgcn_tensor_load_to_lds(g0, g1, z4, z4, z8, 0);
#else
  __builtin_amdgcn_tensor_load_to_lds(g0, g1, z4, z4, 0);
#endif
}

// data_size=2 (4B) | pad_enable | pad_interval | pad_amount
#define TDM_PAD40 (0x00120000u | (4u << 22) | (7u << 25))  // +8 dw every 32 dw
#define TDM_PAD68 (0x00120000u | (5u << 22) | (3u << 25))  // +4 dw every 64 dw
#define TDM_PAD264 (0x00120000u | (7u << 22) | (7u << 25)) // +8 dw every 256 dw

__global__ __launch_bounds__(256)
void mha_fused_kernel(const float* __restrict__ x,
                      const float* __restrict__ Win,
                      const float* __restrict__ bin,
                      const float* __restrict__ Wq,
                      const float* __restrict__ Wk,
                      const float* __restrict__ Wv,
                      const float* __restrict__ Wout,
                      const float* __restrict__ bout,
                      const float* __restrict__ gamma,
                      const float* __restrict__ beta,
                      float* __restrict__ out) {
  __shared__ float lds[LDS_F];
  float* s_ho = lds + OFF_HO;
  float* s_q  = lds + OFF_Q;
  float* s_k  = lds + OFF_K;
  float* s_v  = lds + OFF_V;
  float* s_sc = lds + OFF_SC;
  float* s_C  = lds + OFF_C;
  float* s_p1 = lds + OFF_P1;
  float* s_p2 = lds + OFF_P2;
  float* s_mu = lds + OFF_MU;
  float* s_rs = lds + OFF_RS;

  const int b    = blockIdx.x;
  const int tid  = threadIdx.x;
  const int wave = tid >> 5;
  const int lane = tid & 31;
  const int l16  = lane & 15;
  const int grp  = lane >> 4;

  const float* xb = x + (size_t)b * NC * NT;
  float*       yb = out + (size_t)b * NC * NT;

  const uint32_t ldsC = (uint32_t)(uintptr_t)s_C;   // LDS aperture: low 32b = offset
  const uint32_t ldsB = (uint32_t)(uintptr_t)s_sc;

  // ---- P0: async TDM of Wq/Wk/Wv (-> region B) and first two P1 chunks ----
  if (wave == 0) {
    tdm_load_2d_f32(ldsB + 0u * (ND * WQS * 4u), Wq, ND, ND, ND, ND, ND, TDM_PAD40);
    tdm_load_2d_f32(ldsB + 1u * (ND * WQS * 4u), Wk, ND, ND, ND, ND, ND, TDM_PAD40);
    tdm_load_2d_f32(ldsB + 2u * (ND * WQS * 4u), Wv, ND, ND, ND, ND, ND, TDM_PAD40);
    // chunk 0 and 1 of x (64x64 tile of 64x512) and W_in (64x32 tile of 512x32)
    tdm_load_2d_f32(ldsC + XB0 * 4u, xb, NT, NC, NT, 64, NC, TDM_PAD68);
    tdm_load_2d_f32(ldsC + WB0 * 4u, Win, ND, NT, ND, ND, 64, TDM_PAD40);
    tdm_load_2d_f32(ldsC + XB1 * 4u, xb + 64, NT, NC, NT, 64, NC, TDM_PAD68);
    tdm_load_2d_f32(ldsC + WB1 * 4u, Win + 64 * ND, ND, NT, ND, ND, 64, TDM_PAD40);
  }

  // ---------------- P1: h = x_b @ W_in + b_in, K streamed via TDM ----------------
  {
    const int m0 = (wave >> 1) * 16;
    const int n0 = (wave & 1) * 16;
    v8f acc = {};
    for (int c = 0; c < 8; ++c) {
      if (wave == 0) {
        if (c < 7) __builtin_amdgcn_s_wait_tensorcnt(1);  // chunk c landed
        else       __builtin_amdgcn_s_wait_tensorcnt(0);
      }
      __syncthreads();
      const float* xc = s_C + ((c & 1) ? XB1 : XB0);
      const float* wc = s_C + ((c & 1) ? WB1 : WB0);
#pragma unroll
      for (int kk = 0; kk < 64; kk += 4) {
        v2f a = *(const v2f*)&xc[(m0 + l16) * XCS + kk + 2 * grp];
        v2f bb;
        bb.x = wc[(kk + 2 * grp + 0) * WQS + n0 + l16];
        bb.y = wc[(kk + 2 * grp + 1) * WQS + n0 + l16];
        acc = wmma_f32(a, bb, acc);
      }
      __syncthreads();   // everyone done reading buffer (c&1)
      if (wave == 0 && c + 2 < 8) {
        const int cn = c + 2;
        tdm_load_2d_f32(ldsC + ((cn & 1) ? XB1 : XB0) * 4u, xb + cn * 64,
                        NT, NC, NT, 64, NC, TDM_PAD68);
        tdm_load_2d_f32(ldsC + ((cn & 1) ? WB1 : WB0) * 4u, Win + cn * 64 * ND,
                        ND, NT, ND, ND, 64, TDM_PAD40);
      }
    }
    const float bi = bin[n0 + l16];
#pragma unroll
    for (int r = 0; r < 8; ++r)
      s_ho[(m0 + r + 8 * grp) * HS + n0 + l16] = acc[r] + bi;
  }
  // region C free: start DMA of both W_out halves (overlaps P2+P3)
  if (wave == 0) {
    tdm_load_2d_f32(ldsC + WO0 * 4u, Wout,       NT, ND, NT, 256, ND, TDM_PAD264);
    tdm_load_2d_f32(ldsC + WO1 * 4u, Wout + 256, NT, ND, NT, 256, ND, TDM_PAD264);
  }
  __syncthreads();

  // ---------------- P2: q/k/v = h @ W_{q,k,v} (weights TDM-staged) ----------------
  {
    float* dst[3] = {s_q, s_k, s_v};
    const int m0 = (wave >> 1) * 16;
    const int n0 = (wave & 1) * 16;
    for (int mi = 0; mi < 3; ++mi) {
      const float* W = &s_sc[mi * ND * WQS];
      v8f acc = {};
#pragma unroll
      for (int kk = 0; kk < ND; kk += 4) {
        v2f a = *(const v2f*)&s_ho[(m0 + l16) * HS + kk + 2 * grp];
        v2f bb;
        bb.x = W[(kk + 2 * grp + 0) * WQS + n0 + l16];
        bb.y = W[(kk + 2 * grp + 1) * WQS + n0 + l16];
        acc = wmma_f32(a, bb, acc);
      }
      float* d = dst[mi];
#pragma unroll
      for (int r = 0; r < 8; ++r)
        d[(m0 + r + 8 * grp) * HS + n0 + l16] = acc[r];
    }
  }
  __syncthreads();

  // ---------------- P3: per-head attention over channels ----------------
  const float scale = 0.3535533905932738f;  // 1/sqrt(DK)
  for (int hh = 0; hh < NH; ++hh) {
    for (int jt = 0; jt < 2; ++jt) {        // scores = q_h @ k_h^T
      const int t  = wave + 8 * jt;
      const int m0 = (t >> 2) * 16;
      const int n0 = (t & 3) * 16;
      v8f acc = {};
#pragma unroll
      for (int kk = 0; kk < NDK; kk += 4) {
        v2f a  = *(const v2f*)&s_q[(m0 + l16) * HS + hh * NDK + kk + 2 * grp];
        v2f bb = *(const v2f*)&s_k[(n0 + l16) * HS + hh * NDK + kk + 2 * grp];
        acc = wmma_f32(a, bb, acc);
      }
#pragma unroll
      for (int r = 0; r < 8; ++r)
        s_sc[(m0 + r + 8 * grp) * SS + n0 + l16] = acc[r];
    }
    __syncthreads();

    if (tid < NC) {                         // softmax, one row per thread
      float* row = &s_sc[tid * SS];
      float mx = -1e30f;
      for (int i = 0; i < NC; ++i) mx = fmaxf(mx, row[i]);
      mx *= scale;
      float sum = 0.f;
      for (int i = 0; i < NC; ++i) {
        float e = __expf(row[i] * scale - mx);
        row[i] = e;
        sum += e;
      }
      const float inv = 1.0f / sum;
      for (int i = 0; i < NC; ++i) row[i] *= inv;
    }
    __syncthreads();

    if (wave < 4) {                         // o_h = attn @ v_h (N zero-padded)
      const int m0 = wave * 16;
      v8f acc = {};
      for (int kk = 0; kk < NC; kk += 4) {
        v2f a = *(const v2f*)&s_sc[(m0 + l16) * SS + kk + 2 * grp];
        v2f bb;
        if (l16 < NDK) {
          bb.x = s_v[(kk + 2 * grp + 0) * HS + hh * NDK + l16];
          bb.y = s_v[(kk + 2 * grp + 1) * HS + hh * NDK + l16];
        } else {
          bb.x = 0.f; bb.y = 0.f;
        }
        acc = wmma_f32(a, bb, acc);
      }
      if (l16 < NDK) {
#pragma unroll
        for (int r = 0; r < 8; ++r)
          s_ho[(m0 + r + 8 * grp) * HS + hh * NDK + l16] = acc[r];
      }
    }
    __syncthreads();
  }

  // W_out halves must be resident now
  if (wave == 0) __builtin_amdgcn_s_wait_tensorcnt(0);
  __syncthreads();

  // ---------------- P4: y = o @ W_out + b_out + x; then layernorm ----------------
  {
    float rs1[8], rs2[8];
#pragma unroll
    for (int r = 0; r < 8; ++r) { rs1[r] = 0.f; rs2[r] = 0.f; }
    const int m0 = (wave & 3) * 16;  // m-tile fixed per wave (deterministic reduce)
    for (int jt = 0; jt < 16; ++jt) {
      const int t  = wave + 8 * jt;
      const int n0 = (t >> 2) * 16;
      const float* wo = s_C + ((n0 >> 8) ? WO1 : WO0);  // staged half
      const int nl = n0 & 255;
      v8f acc = {};
#pragma unroll
      for (int kk = 0; kk < ND; kk += 4) {
        v2f a = *(const v2f*)&s_ho[(m0 + l16) * HS + kk + 2 * grp];
        v2f bb;
        bb.x = wo[(kk + 2 * grp + 0) * WOS + nl + l16];
        bb.y = wo[(kk + 2 * grp + 1) * WOS + nl + l16];
        acc = wmma_f32(a, bb, acc);
      }
      const float bo = bout[n0 + l16];
#pragma unroll
      for (int r = 0; r < 8; ++r) {
        const int row = m0 + r + 8 * grp;
        const int col = n0 + l16;
        float y = acc[r] + bo + xb[row * NT + col];
        yb[row * NT + col] = y;
        float a1 = y, a2 = y * y;
        a1 += __shfl_xor(a1, 1);  a2 += __shfl_xor(a2, 1);
        a1 += __shfl_xor(a1, 2);  a2 += __shfl_xor(a2, 2);
        a1 += __shfl_xor(a1, 4);  a2 += __shfl_xor(a2, 4);
        a1 += __shfl_xor(a1, 8);  a2 += __shfl_xor(a2, 8);
        rs1[r] += a1; rs2[r] += a2;
      }
    }
    const int halfIdx = wave >> 2;
    if (l16 == 0) {
#pragma unroll
      for (int r = 0; r < 8; ++r) {
        const int row = m0 + r + 8 * grp;
        s_p1[row * 2 + halfIdx] = rs1[r];
        s_p2[row * 2 + halfIdx] = rs2[r];
      }
    }
  }
  __threadfence();
  __syncthreads();

  if (tid < NC) {
    const float s1 = s_p1[tid * 2] + s_p1[tid * 2 + 1];
    const float s2 = s_p2[tid * 2] + s_p2[tid * 2 + 1];
    const float mu = s1 * (1.0f / NT);
    const float var = s2 * (1.0f / NT) - mu * mu;
    s_mu[tid] = mu;
    s_rs[tid] = rsqrtf(var + EPS);
  }
  __syncthreads();

  {
    float4* y4 = (float4*)yb;
    const float4* g4 = (const float4*)gamma;
    const float4* b4 = (const float4*)beta;
    for (int i = tid; i < (NC * NT) / 4; i += 256) {
      const int row = i >> 7;
      const int c4  = i & 127;
      float4 yv = y4[i];
      const float4 gv = g4[c4];
      const float4 bv = b4[c4];
      const float mu = s_mu[row], rs = s_rs[row];
      yv.x = (yv.x - mu) * rs * gv.x + bv.x;
      yv.y = (yv.y - mu) * rs * gv.y + bv.y;
      yv.z = (yv.z - mu) * rs * gv.z + bv.z;
      yv.w = (yv.w - mu) * rs * gv.w + bv.w;
      y4[i] = yv;
    }
  }
}

extern "C" void kernel_launch(void* const* d_in, const int* in_sizes, int n_in,
                              void* d_out, int out_size, void* d_ws, size_t ws_size,
                              hipStream_t stream) {
  (void)in_sizes; (void)n_in; (void)out_size; (void)d_ws; (void)ws_size;
  const float* x     = (const float*)d_in[0];
  const float* Win   = (const float*)d_in[1];
  const float* bin   = (const float*)d_in[2];
  const float* Wq    = (const float*)d_in[3];
  const float* Wk    = (const float*)d_in[4];
  const float* Wv    = (const float*)d_in[5];
  const float* Wout  = (const float*)d_in[6];
  const float* bout  = (const float*)d_in[7];
  const float* gamma = (const float*)d_in[8];
  const float* beta  = (const float*)d_in[9];
  float* out = (float*)d_out;

  mha_fused_kernel<<<dim3(NB), dim3(256), 0, stream>>>(
      x, Win, bin, Wq, Wk, Wv, Wout, bout, gamma, beta, out);
}